// ZSSR_INV_35536559407194
// MI455X (gfx1250) — compile-verified
//
#include <hip/hip_runtime.h>

typedef __attribute__((ext_vector_type(2))) float v2f;
typedef __attribute__((ext_vector_type(8))) float v8f;

#define HP    130
#define WP    130
#define PIX   (HP * WP)          // 16900
#define MID   64
#define RED   16
#define K7    7
#define KK    49
#define GROUPS 4
#define GCH   16
#define SPANM (KK * GROUPS)      // 196
#define POUT  (128 * 128)        // conv_out spatial, 16384 = 1024 * 16

// ---------------------------------------------------------------------------
// conv_in: (1,3,128,128) -> (64,130,130), 3x3 pad=2. One thread per (c, pixel).
// Kept on VALU: only ~29M MACs and padding-dominated (boundary masks would
// cost more than the MACs on the WMMA path).
// ---------------------------------------------------------------------------
__global__ void zssr_conv_in(const float* __restrict__ in,
                             const float* __restrict__ wgt,
                             const float* __restrict__ bias,
                             float* __restrict__ X) {
    int idx = blockIdx.x * blockDim.x + threadIdx.x;
    if (idx >= MID * PIX) return;
    int c = idx / PIX;
    int p = idx - c * PIX;
    int h = p / WP, w = p - h * WP;
    float acc = bias[c];
#pragma unroll
    for (int ci = 0; ci < 3; ++ci) {
#pragma unroll
        for (int kh = 0; kh < 3; ++kh) {
            int hs = h - 2 + kh;
            if ((unsigned)hs >= 128u) continue;
#pragma unroll
            for (int kw = 0; kw < 3; ++kw) {
                int ws = w - 2 + kw;
                if ((unsigned)ws >= 128u) continue;
                acc += in[(ci * 128 + hs) * 128 + ws] *
                       wgt[((c * 3 + ci) * 3 + kh) * 3 + kw];
            }
        }
    }
    X[idx] = acc;
}

// ---------------------------------------------------------------------------
// red GEMM + BN + ReLU:  T(16,P) = relu(gamma * (Wred(16,64) @ X(64,P)) + beta)
// One wave per 16-pixel tile; K=64 -> 16x v_wmma_f32_16x16x4_f32.
// ---------------------------------------------------------------------------
__global__ void zssr_red_gemm(const float* __restrict__ X,
                              const float* __restrict__ Wred,   // (16,64)
                              const float* __restrict__ gamma,  // (16)
                              const float* __restrict__ beta,   // (16)
                              float* __restrict__ T) {
    const int wavesPerBlk = blockDim.x >> 5;
    int wave = blockIdx.x * wavesPerBlk + (threadIdx.x >> 5);
    int lane = threadIdx.x & 31;
    int p0 = wave * 16;
    if (p0 >= PIX) return;                 // wave-uniform exit, EXEC stays full

    int half = lane >> 4;                  // 0: K-lo half, 1: K-hi half
    int l16  = lane & 15;
    int p    = p0 + l16;
    int pc   = p < PIX ? p : PIX - 1;      // clamp (select, no divergence)

    v8f acc = {};
#pragma unroll
    for (int kb = 0; kb < 64; kb += 4) {
        int k0 = kb + 2 * half;
        v2f a, b;
        a.x = Wred[l16 * 64 + k0];
        a.y = Wred[l16 * 64 + k0 + 1];
        b.x = X[k0 * PIX + pc];
        b.y = X[(k0 + 1) * PIX + pc];
        acc = __builtin_amdgcn_wmma_f32_16x16x4_f32(
                  false, a, false, b, (short)0, acc, false, false);
    }
    if (p < PIX) {
#pragma unroll
        for (int r = 0; r < 8; ++r) {
            int m = r + 8 * half;          // C/D layout: row = r + 8*laneHalf
            float v = gamma[m] * acc[r] + beta[m];
            T[m * PIX + p] = v > 0.f ? v : 0.f;
        }
    }
}

// ---------------------------------------------------------------------------
// span GEMM + bias: Wk(196,P) = Wspan(196,16) @ T(16,P) + b
// grid.y = 13 M-tiles (196 -> 13x16, rows >=196 padded), K=16 -> 4 WMMAs.
// ---------------------------------------------------------------------------
__global__ void zssr_span_gemm(const float* __restrict__ T,
                               const float* __restrict__ Wspan,  // (196,16)
                               const float* __restrict__ Bspan,  // (196)
                               float* __restrict__ Wk) {
    const int wavesPerBlk = blockDim.x >> 5;
    int wave = blockIdx.x * wavesPerBlk + (threadIdx.x >> 5);
    int mt   = blockIdx.y;
    int lane = threadIdx.x & 31;
    int p0 = wave * 16;
    if (p0 >= PIX) return;

    int half = lane >> 4;
    int l16  = lane & 15;
    int p    = p0 + l16;
    int pc   = p < PIX ? p : PIX - 1;
    int mrow = mt * 16 + l16;
    int mcl  = mrow < SPANM ? mrow : SPANM - 1;   // clamp weight row for loads

    v8f acc = {};
#pragma unroll
    for (int kb = 0; kb < 16; kb += 4) {
        int k0 = kb + 2 * half;
        v2f a, b;
        a.x = Wspan[mcl * 16 + k0];
        a.y = Wspan[mcl * 16 + k0 + 1];
        b.x = T[k0 * PIX + pc];
        b.y = T[(k0 + 1) * PIX + pc];
        acc = __builtin_amdgcn_wmma_f32_16x16x4_f32(
                  false, a, false, b, (short)0, acc, false, false);
    }
    if (p < PIX) {
#pragma unroll
        for (int r = 0; r < 8; ++r) {
            int m = mt * 16 + r + 8 * half;
            if (m < SPANM)
                Wk[m * PIX + p] = acc[r] + Bspan[m];
        }
    }
}

// ---------------------------------------------------------------------------
// involution + outer ReLU: per-pixel dynamic 7x7 conv, grouped (4 groups x 16ch).
// One thread per (channel, pixel). Weights differ per output column -> batched
// matvec, inherently non-WMMA; VALU FMA.
// ---------------------------------------------------------------------------
__global__ void zssr_involution(const float* __restrict__ X,
                                const float* __restrict__ Wk,   // (196,P)
                                float* __restrict__ Y) {
    int idx = blockIdx.x * blockDim.x + threadIdx.x;
    if (idx >= MID * PIX) return;
    int c = idx / PIX;
    int p = idx - c * PIX;
    int h = p / WP, w = p - h * WP;
    int g = c >> 4;
    const float* xc = X + c * PIX;
    const float* wg = Wk + g * KK * PIX + p;
    float acc = 0.f;
#pragma unroll
    for (int kh = 0; kh < K7; ++kh) {
        int hs = h + kh - 3;
        if ((unsigned)hs >= (unsigned)HP) continue;
#pragma unroll
        for (int kw = 0; kw < K7; ++kw) {
            int ws = w + kw - 3;
            if ((unsigned)ws >= (unsigned)WP) continue;
            acc += wg[(kh * K7 + kw) * PIX] * xc[hs * WP + ws];
        }
    }
    Y[idx] = acc > 0.f ? acc : 0.f;
}

// ---------------------------------------------------------------------------
// conv_out as im2col WMMA GEMM, fused with PixelShuffle(2).
//   pixel(16pad x POUT) = Wout(16pad x 576) @ im2col(576 x POUT) + bias
// Valid conv (130->128): every tap in-bounds, so the B operand needs no edge
// masks. K ordered as tap-major (9 taps x 64 cin) -> 144 WMMA steps/wave.
// Rows 12..15 get zero weights (lane select) and are never stored.
// POUT = 16384 = 1024 exact 16-pixel tiles -> no pixel clamp either.
// ---------------------------------------------------------------------------
__global__ void zssr_conv_out_wmma(const float* __restrict__ X,     // (64,130,130)
                                   const float* __restrict__ wgt,   // (12,64,3,3)
                                   const float* __restrict__ bias,  // (12)
                                   float* __restrict__ out) {       // (3,256,256)
    const int wavesPerBlk = blockDim.x >> 5;
    int wave = blockIdx.x * wavesPerBlk + (threadIdx.x >> 5);
    int lane = threadIdx.x & 31;
    int p0 = wave * 16;
    if (p0 >= POUT) return;                // wave-uniform

    int half = lane >> 4;
    int l16  = lane & 15;
    int p    = p0 + l16;                   // this lane's output column (pixel)
    int h    = p >> 7, w = p & 127;

    int  m      = l16;                     // A-operand row this lane feeds
    int  mcl    = m < 12 ? m : 11;         // clamp for load safety
    bool mvalid = m < 12;

    v8f acc = {};
    for (int tap = 0; tap < 9; ++tap) {
        int kh = tap / 3, kw = tap - kh * 3;
        int srcoff = (h + kh) * WP + (w + kw);   // always in-bounds (valid conv)
#pragma unroll
        for (int cinb = 0; cinb < MID; cinb += 4) {
            int cin0 = cinb + 2 * half;
            v2f a, b;
            float ax = wgt[(mcl * MID + cin0) * 9 + tap];
            float ay = wgt[(mcl * MID + cin0 + 1) * 9 + tap];
            a.x = mvalid ? ax : 0.f;       // zero-pad rows 12..15
            a.y = mvalid ? ay : 0.f;
            b.x = X[cin0 * PIX + srcoff];
            b.y = X[(cin0 + 1) * PIX + srcoff];
            acc = __builtin_amdgcn_wmma_f32_16x16x4_f32(
                      false, a, false, b, (short)0, acc, false, false);
        }
    }
    // store with fused PixelShuffle: pixel[mm, h, w] -> out[mm>>2, 2h+i, 2w+j]
#pragma unroll
    for (int r = 0; r < 8; ++r) {
        int mm = r + 8 * half;
        if (mm < 12) {
            float v = acc[r] + bias[mm];
            int c = mm >> 2, ij = mm & 3;
            int i = ij >> 1, j = ij & 1;
            out[c * 65536 + (2 * h + i) * 256 + (2 * w + j)] = v;
        }
    }
}

// ---------------------------------------------------------------------------
extern "C" void kernel_launch(void* const* d_in, const int* in_sizes, int n_in,
                              void* d_out, int out_size, void* d_ws, size_t ws_size,
                              hipStream_t stream) {
    const float* input    = (const float*)d_in[0];  // (1,3,128,128)
    const float* conv_in_w = (const float*)d_in[1]; // (64,3,3,3)
    const float* conv_in_b = (const float*)d_in[2]; // (64)
    const float* red_w    = (const float*)d_in[3];  // (6,16,64)
    const float* bn_gamma = (const float*)d_in[4];  // (6,16)
    const float* bn_beta  = (const float*)d_in[5];  // (6,16)
    const float* span_w   = (const float*)d_in[6];  // (6,196,16)
    const float* span_b   = (const float*)d_in[7];  // (6,196)
    const float* conv_out_w = (const float*)d_in[8]; // (12,64,3,3)
    const float* conv_out_b = (const float*)d_in[9]; // (12)
    float* out = (float*)d_out;                      // (1,3,256,256)

    float* ws = (float*)d_ws;
    float* X0 = ws;                         // 64*PIX
    float* X1 = X0 + MID * PIX;             // 64*PIX
    float* T  = X1 + MID * PIX;             // 16*PIX
    float* Wk = T + RED * PIX;              // 196*PIX   (~23 MB total)

    const int nElem = MID * PIX;            // 1,081,600
    const int blk = 256;

    // 1. input conv
    zssr_conv_in<<<(nElem + blk - 1) / blk, blk, 0, stream>>>(
        input, conv_in_w, conv_in_b, X0);

    // 2. six involution blocks (ping-pong X0 <-> X1)
    const int pTiles = (PIX + 15) / 16;                 // 1057 waves
    const int wavesPerBlk = blk / 32;                   // 8
    const int gemmBlocks = (pTiles + wavesPerBlk - 1) / wavesPerBlk;

    float* xa = X0;
    float* xb = X1;
    for (int l = 0; l < 6; ++l) {
        zssr_red_gemm<<<gemmBlocks, blk, 0, stream>>>(
            xa, red_w + l * RED * MID,
            bn_gamma + l * RED, bn_beta + l * RED, T);

        dim3 sg(gemmBlocks, (SPANM + 15) / 16);         // x: pixel tiles, y: 13
        zssr_span_gemm<<<sg, blk, 0, stream>>>(
            T, span_w + l * SPANM * RED, span_b + l * SPANM, Wk);

        zssr_involution<<<(nElem + blk - 1) / blk, blk, 0, stream>>>(
            xa, Wk, xb);

        float* tmp = xa; xa = xb; xb = tmp;
    }

    // 3. output conv (WMMA im2col GEMM) + fused pixel shuffle
    const int outTiles = POUT / 16;                     // 1024, exact
    const int outBlocks = (outTiles + wavesPerBlk - 1) / wavesPerBlk;
    zssr_conv_out_wmma<<<outBlocks, blk, 0, stream>>>(
        xa, conv_out_w, conv_out_b, out);
}